// ClassAttn_64330020159871
// MI455X (gfx1250) — compile-verified
//
#include <hip/hip_runtime.h>
#include <math.h>

// Problem constants (reference: B=32, N=4096, C=768, H=12, D=64)
#define Csz   768
#define NTOK  4096
#define NH    12
#define DH    64
#define BSZ   32
#define SCALE 0.125f   // 64^-0.5

typedef __attribute__((ext_vector_type(2))) float v2f;
typedef __attribute__((ext_vector_type(8))) float v8f;

// ---------------------------------------------------------------------------
// CDNA5 async global->LDS copy (tracked by ASYNCcnt, bypasses VGPRs)
// ---------------------------------------------------------------------------
__device__ __forceinline__ void async_ld_b128(unsigned lds_off, const float* g) {
  asm volatile("global_load_async_to_lds_b128 %0, %1, off"
               :: "v"(lds_off), "v"(g) : "memory");
}
__device__ __forceinline__ void wait_async0() {
  asm volatile("s_wait_asynccnt 0x0" ::: "memory");
}
// generic shared pointer -> wave-relative LDS byte offset (flat LDS aperture
// keeps the offset in addr[31:0])
__device__ __forceinline__ unsigned lds_off_of(const void* p) {
  return (unsigned)(uintptr_t)p;
}

// ---------------------------------------------------------------------------
// K0a: q[b,j] = (x[b,0,:] @ Wq + bq)[j] * SCALE           (B*C = 24576 outputs)
// ---------------------------------------------------------------------------
__global__ void k_qproj(const float* __restrict__ x, const float* __restrict__ Wq,
                        const float* __restrict__ bq, float* __restrict__ q) {
  int g = blockIdx.x * blockDim.x + threadIdx.x;
  if (g >= BSZ * Csz) return;
  int b = g / Csz, j = g % Csz;
  const float* xr = x + (size_t)b * NTOK * Csz;   // row n=0
  float acc = bq[j];
  for (int c = 0; c < Csz; ++c) acc += xr[c] * Wq[(size_t)c * Csz + j];
  q[g] = acc * SCALE;
}

// ---------------------------------------------------------------------------
// K0b: wt[b,h,c] = sum_d Wk[c, h*64+d] * q[b, h*64+d]     (B*H*C outputs)
// ---------------------------------------------------------------------------
__global__ void k_wtilde(const float* __restrict__ Wk, const float* __restrict__ q,
                         float* __restrict__ wt) {
  int g = blockIdx.x * blockDim.x + threadIdx.x;
  if (g >= BSZ * NH * Csz) return;
  int b = g / (NH * Csz);
  int r = g % (NH * Csz);
  int h = r / Csz, c = r % Csz;
  const float* wkr = Wk + (size_t)c * Csz + h * DH;
  const float* qr  = q + (size_t)b * Csz + h * DH;
  float acc = 0.f;
  for (int d = 0; d < DH; ++d) acc += wkr[d] * qr[d];
  wt[g] = acc;
}

// ---------------------------------------------------------------------------
// K0c: lbias[b,h] = sum_d bk[h*64+d] * q[b, h*64+d]
// ---------------------------------------------------------------------------
__global__ void k_lbias(const float* __restrict__ bk, const float* __restrict__ q,
                        float* __restrict__ lbias) {
  int g = blockIdx.x * blockDim.x + threadIdx.x;
  if (g >= BSZ * NH) return;
  int b = g / NH, h = g % NH;
  const float* qr = q + (size_t)b * Csz + h * DH;
  float acc = 0.f;
  for (int d = 0; d < DH; ++d) acc += bk[h * DH + d] * qr[d];
  lbias[g] = acc;
}

// ---------------------------------------------------------------------------
// K1: logits[b,h,n] = x[b,n,:]·wt[b,h,:] + lbias[b,h]  via v_wmma_f32_16x16x4_f32
// grid = (row_tiles=32, b=32); block = 256 (8 waves); wave w -> rows w*16..+15
// x staged via double-buffered async global->LDS copies (64-c chunks).
// LDS buffers selected by INTEGER offsets from the shared base so addrspace(3)
// is preserved (ds_load, not flat_load).
// ---------------------------------------------------------------------------
#define WT_STRIDE 772            // 768 + 4 pad (bank-conflict-free b64 reads)
#define XS_STRIDE 68             // 64 + 4 pad
#define X1BUF     (128 * XS_STRIDE)
__global__ void k_logits(const float* __restrict__ x, const float* __restrict__ wt,
                         const float* __restrict__ lbias, float* __restrict__ attn_out) {
  extern __shared__ float smem[];
  // layout: [0, 16*WT_STRIDE) = wts ; then two 128x68 x buffers
  const int tid   = threadIdx.x;
  const int b     = blockIdx.y;
  const int rbase = blockIdx.x * 128;

  // stage w-tilde for this batch; pad heads 12..15 with zeros
  for (int i = tid; i < 16 * Csz; i += 256) {
    int h = i / Csz, c = i % Csz;
    smem[h * WT_STRIDE + c] = (h < NH) ? wt[((size_t)b * NH + h) * Csz + c] : 0.f;
  }

  const float* xrowbase = x + ((size_t)b * NTOK + rbase) * Csz;
  // async-issue one 128x64 x chunk into buffer p (0/1)
  auto issue_x = [&](int p, int c0) {
    const int boff = 16 * WT_STRIDE + p * X1BUF;
    for (int i = tid; i < 128 * 16; i += 256) {
      int row = i >> 4, c4 = i & 15;
      async_ld_b128(lds_off_of(smem + boff + row * XS_STRIDE + c4 * 4),
                    xrowbase + (size_t)row * Csz + c0 + c4 * 4);
    }
  };
  issue_x(0, 0);

  const int lane  = tid & 31;
  const int wave  = tid >> 5;
  const int mrow  = lane & 15;                 // M (A) / N (B) index of this lane
  const int khalf = (lane < 16) ? 0 : 2;       // K split per ISA 16x4 f32 layout
  v8f acc = {0.f, 0.f, 0.f, 0.f, 0.f, 0.f, 0.f, 0.f};

  for (int ci = 0; ci < Csz / 64; ++ci) {
    const int c0 = ci * 64;
    wait_async0();          // own chunk-ci loads done
    __syncthreads();        // all waves' chunk-ci LDS writes visible (+wt stage)
    if (ci + 1 < Csz / 64) issue_x((ci + 1) & 1, c0 + 64);   // prefetch
    // integer-offset GEP from smem keeps these as ds_load
    const float* arow = smem + 16 * WT_STRIDE + (ci & 1) * X1BUF
                             + (wave * 16 + mrow) * XS_STRIDE;
    const float* brow = smem + mrow * WT_STRIDE + c0;
#pragma unroll
    for (int k = 0; k < 64; k += 4) {
      v2f a, bm;
      a.x  = arow[k + khalf];  a.y  = arow[k + khalf + 1];
      bm.x = brow[k + khalf];  bm.y = brow[k + khalf + 1];
      acc = __builtin_amdgcn_wmma_f32_16x16x4_f32(false, a, false, bm,
                                                  (short)0, acc, false, false);
    }
  }

  // D layout: vgpr r, lanes 0-15 -> M=r, N=lane ; lanes 16-31 -> M=r+8, N=lane-16
  int head = lane & 15;
  if (head < NH) {
    float lb  = lbias[b * NH + head];
    int  radd = (lane < 16) ? 0 : 8;
#pragma unroll
    for (int r = 0; r < 8; ++r) {
      int nrow = rbase + wave * 16 + r + radd;
      attn_out[((size_t)b * NH + head) * NTOK + nrow] = acc[r] + lb;
    }
  }
}

// ---------------------------------------------------------------------------
// K2: stable softmax over keys, in place in the attn output region
// grid = B*H blocks of 256 threads
// ---------------------------------------------------------------------------
__global__ void k_softmax(float* __restrict__ attn) {
  __shared__ float red[256];
  const int tid = threadIdx.x;
  float* row = attn + (size_t)blockIdx.x * NTOK;
  float m = -1e30f;
  for (int i = tid; i < NTOK; i += 256) m = fmaxf(m, row[i]);
  red[tid] = m; __syncthreads();
  for (int s = 128; s > 0; s >>= 1) {
    if (tid < s) red[tid] = fmaxf(red[tid], red[tid + s]);
    __syncthreads();
  }
  m = red[0]; __syncthreads();
  float z = 0.f;
  for (int i = tid; i < NTOK; i += 256) z += __expf(row[i] - m);
  red[tid] = z; __syncthreads();
  for (int s = 128; s > 0; s >>= 1) {
    if (tid < s) red[tid] += red[tid + s];
    __syncthreads();
  }
  float inv = 1.f / red[0];
  for (int i = tid; i < NTOK; i += 256) row[i] = __expf(row[i] - m) * inv;
}

// ---------------------------------------------------------------------------
// K3: ypart[seg,b,h,c] = sum_{n in seg} attn[b,h,n] * x[b,n,c]  via WMMA
// grid = (cblk=6, seg=4, b=32); block = 256; wave w -> c cols w*16..+15 of 128
// A = attn tile (16 head rows x 64 n), B = x tile (64 n x 128 c)
// both tiles double-buffered via async global->LDS; integer-offset buffer select
// ---------------------------------------------------------------------------
#define X3_STRIDE 132            // 128 + 4 pad
#define A3_STRIDE 68             // 64 + 4 pad
#define X3BUF     (64 * X3_STRIDE)
#define A3OFF     (2 * X3BUF)    // attn buffers start after the two x buffers
#define A3BUF     (16 * A3_STRIDE)
__global__ void k_yacc(const float* __restrict__ x, const float* __restrict__ attn,
                       float* __restrict__ ypart) {
  extern __shared__ float smem[];
  const int tid  = threadIdx.x;
  const int cblk = blockIdx.x;           // 0..5
  const int seg  = blockIdx.y;           // 0..3
  const int b    = blockIdx.z;
  const int cbase    = cblk * 128;
  const int nsegbase = seg * (NTOK / 4);

  // zero-pad head rows 12..15 of both attn buffers once (never refreshed)
  for (int i = tid; i < 4 * 64 * 2; i += 256) {
    int buf = i >> 8, r = i & 255;
    smem[A3OFF + buf * A3BUF + (NH + (r >> 6)) * A3_STRIDE + (r & 63)] = 0.f;
  }

  const float* xbase = x + (size_t)b * NTOK * Csz + cbase;
  const float* abase = attn + (size_t)b * NH * NTOK + nsegbase;
  // async-issue the (64 n x 128 c) x tile and (12 h x 64 n) attn tile into buf p
  auto issue = [&](int p, int n0) {
    const int xoff = p * X3BUF;
    for (int i = tid; i < 64 * 32; i += 256) {
      int n = i >> 5, c4 = i & 31;
      async_ld_b128(lds_off_of(smem + xoff + n * X3_STRIDE + c4 * 4),
                    xbase + (size_t)(n0 + n) * Csz + c4 * 4);
    }
    if (tid < NH * 16) {
      int h = tid >> 4, j4 = tid & 15;
      async_ld_b128(lds_off_of(smem + A3OFF + p * A3BUF + h * A3_STRIDE + j4 * 4),
                    abase + (size_t)h * NTOK + n0 + j4 * 4);
    }
  };
  issue(0, 0);

  const int lane  = tid & 31;
  const int wave  = tid >> 5;
  const int khalf = (lane < 16) ? 0 : 2;
  v8f acc = {0.f, 0.f, 0.f, 0.f, 0.f, 0.f, 0.f, 0.f};

  for (int ni = 0; ni < (NTOK / 4) / 64; ++ni) {
    const int n0 = ni * 64;
    wait_async0();
    __syncthreads();
    if (ni + 1 < (NTOK / 4) / 64) issue((ni + 1) & 1, n0 + 64);   // prefetch
    const int p = ni & 1;
    const float* arow = smem + A3OFF + p * A3BUF + (lane & 15) * A3_STRIDE;
    const float* xcur = smem + p * X3BUF;
    const int ccol = wave * 16 + (lane & 15);
#pragma unroll
    for (int k = 0; k < 64; k += 4) {
      v2f a, bm;
      a.x  = arow[k + khalf];  a.y = arow[k + khalf + 1];
      bm.x = xcur[(k + khalf) * X3_STRIDE + ccol];
      bm.y = xcur[(k + khalf + 1) * X3_STRIDE + ccol];
      acc = __builtin_amdgcn_wmma_f32_16x16x4_f32(false, a, false, bm,
                                                  (short)0, acc, false, false);
    }
  }

  // D: M = head row, N = c col
  int radd = (lane < 16) ? 0 : 8;
  int ccol = cbase + wave * 16 + (lane & 15);
#pragma unroll
  for (int r = 0; r < 8; ++r) {
    int head = r + radd;
    if (head < NH)
      ypart[(((size_t)seg * BSZ + b) * NH + head) * Csz + ccol] = acc[r];
  }
}

// ---------------------------------------------------------------------------
// K4a: xc[b, h*64+d] = sum_c ysum[b,h,c] * Wv[c, h*64+d] + bv[...]
// ---------------------------------------------------------------------------
__global__ void k_xcls(const float* __restrict__ ypart, const float* __restrict__ Wv,
                       const float* __restrict__ bv, float* __restrict__ xc) {
  int g = blockIdx.x * blockDim.x + threadIdx.x;
  if (g >= BSZ * Csz) return;
  int b = g / Csz, e = g % Csz, h = e / DH;
  const float* y0 = ypart + (((size_t)0 * BSZ + b) * NH + h) * Csz;
  const float* y1 = ypart + (((size_t)1 * BSZ + b) * NH + h) * Csz;
  const float* y2 = ypart + (((size_t)2 * BSZ + b) * NH + h) * Csz;
  const float* y3 = ypart + (((size_t)3 * BSZ + b) * NH + h) * Csz;
  float acc = bv[e];
  for (int c = 0; c < Csz; ++c) {
    float ys = y0[c] + y1[c] + y2[c] + y3[c];
    acc += ys * Wv[(size_t)c * Csz + e];
  }
  xc[g] = acc;
}

// ---------------------------------------------------------------------------
// K4b: out[b,j] = xc[b,:] @ Wp[:,j] + bp[j]
// ---------------------------------------------------------------------------
__global__ void k_proj(const float* __restrict__ xc, const float* __restrict__ Wp,
                       const float* __restrict__ bp, float* __restrict__ out) {
  int g = blockIdx.x * blockDim.x + threadIdx.x;
  if (g >= BSZ * Csz) return;
  int b = g / Csz, j = g % Csz;
  const float* xr = xc + (size_t)b * Csz;
  float acc = bp[j];
  for (int e = 0; e < Csz; ++e) acc += xr[e] * Wp[(size_t)e * Csz + j];
  out[g] = acc;
}

// ---------------------------------------------------------------------------
extern "C" void kernel_launch(void* const* d_in, const int* in_sizes, int n_in,
                              void* d_out, int out_size, void* d_ws, size_t ws_size,
                              hipStream_t stream) {
  const float* x  = (const float*)d_in[0];
  const float* Wq = (const float*)d_in[1];
  const float* bq = (const float*)d_in[2];
  const float* Wk = (const float*)d_in[3];
  const float* bk = (const float*)d_in[4];
  const float* Wv = (const float*)d_in[5];
  const float* bv = (const float*)d_in[6];
  const float* Wp = (const float*)d_in[7];
  const float* bp = (const float*)d_in[8];

  float* out_xcls = (float*)d_out;                 // B*C = 24576 floats
  float* out_attn = (float*)d_out + BSZ * Csz;     // B*H*N floats

  // workspace layout (floats)
  float* ws    = (float*)d_ws;
  float* q     = ws;                                   // 24576
  float* wt    = q + BSZ * Csz;                        // 294912
  float* lbias = wt + BSZ * NH * Csz;                  // 384
  float* ypart = lbias + BSZ * NH;                     // 4*32*12*768 = 1179648
  float* xc    = ypart + 4 * BSZ * NH * Csz;           // 24576

  // stage 0: q, w-tilde, logit bias
  k_qproj<<<(BSZ * Csz + 255) / 256, 256, 0, stream>>>(x, Wq, bq, q);
  k_wtilde<<<(BSZ * NH * Csz + 255) / 256, 256, 0, stream>>>(Wk, q, wt);
  k_lbias<<<(BSZ * NH + 255) / 256, 256, 0, stream>>>(bk, q, lbias);

  // stage 1: logits via WMMA + async double-buffered staging (reads x once)
  size_t shm1 = (size_t)(16 * WT_STRIDE + 2 * X1BUF) * sizeof(float);
  k_logits<<<dim3(NTOK / 128, BSZ), 256, shm1, stream>>>(x, wt, lbias, out_attn);

  // stage 2: softmax normalize attn in place
  k_softmax<<<BSZ * NH, 256, 0, stream>>>(out_attn);

  // stage 3: y = attn^T-weighted sum of x via WMMA (reads x once more)
  size_t shm3 = (size_t)(2 * X3BUF + 2 * A3BUF) * sizeof(float);
  k_yacc<<<dim3(Csz / 128, 4, BSZ), 256, shm3, stream>>>(x, out_attn, ypart);

  // stage 4: head-slice Wv projection + final Wp projection
  k_xcls<<<(BSZ * Csz + 255) / 256, 256, 0, stream>>>(ypart, Wv, bv, xc);
  k_proj<<<(BSZ * Csz + 255) / 256, 256, 0, stream>>>(xc, Wp, bp, out_xcls);
}